// vectorsExtractor_42460046688734
// MI455X (gfx1250) — compile-verified
//
#include <hip/hip_runtime.h>
#include <math.h>

// Problem constants (from reference): feats [8,256,128,256] f32, y_down [8,128,256] i64
#define B_    8
#define C_    256
#define H_    128
#define W_    256
#define HW    (H_ * W_)          // 32768 pixels per (b,c) plane
#define NPIX  (B_ * HW)          // 262144 total pixels
#define K_    19                 // classes
#define PIX   256                // pixels per workgroup (divides HW)
#define NWG   (NPIX / PIX)       // 1024 workgroups in main kernel
#define KC    (K_ * C_)          // 4864 = per-WG partial-sum tile size
#define PFDIST 4                 // prefetch distance in 32-pixel groups (~512B/plane)

// Two-stage reducer geometry
#define RCHUNK 64                // main-kernel WGs folded per stage-A block
#define NCHUNK (NWG / RCHUNK)    // 16
#define JT     128               // output elements per stage-A block
#define NJT    (KC / JT)         // 38 (exact: 4864 = 38*128)

typedef __attribute__((ext_vector_type(2))) float v2f;
typedef __attribute__((ext_vector_type(8))) float v8f;

// -----------------------------------------------------------------------------
// Kernel 1: per-workgroup partial segment sums via V_WMMA_F32_16X16X4_F32.
//   grid = NWG blocks x 128 threads (4 waves). Wave w owns channels [64w,64w+64)
//   (4 channel groups of 16). All 4 waves share the same PIX pixels, so the
//   per-pixel sum-of-squares completes inside the workgroup, and the 19x256
//   class sums flush as disjoint per-WG partials to d_ws (no global atomics).
//
// WMMA mapping per 4-pixel step, per 16-channel group:
//   A (16x4, MxK)  = feats[ch=cbase+M][pixel lp+K]  -> per-lane global_load_b64
//                    (lanes 0-15: K=0,1 ; lanes 16-31: K=2,3 — ISA A layout)
//   B (4x16, KxN)  = onehot[pixel lp+K][class N]    -> v_cmp/v_cndmask from
//                    labels read directly from global (LOADcnt, not DScnt,
//                    so the fire-and-forget ds_add_f32s are never waited on)
//   C/D (16x16)    = sums[channel M][class N], f32 accumulate (exact f32 math)
//
// Software pipeline: step s+1's A fragments + labels are loaded before step
// s's WMMAs issue (~2x outstanding loads); global_prefetch_b8 runs PFDIST
// groups ahead so demand loads hit L2/WGP$ rather than raw HBM latency.
// -----------------------------------------------------------------------------
__global__ __launch_bounds__(128) void seg_partial_kernel(
    const float* __restrict__ feats,
    const long long* __restrict__ ydown,
    float* __restrict__ ws_sums,   // [NWG][K_][C_]
    float* __restrict__ ws_norm,   // [NWG][K_]
    float* __restrict__ ws_cnt)    // [NWG][K_]
{
    __shared__ float ssqp[PIX * 16];   // per-pixel, per-column x^2 partials
    __shared__ float nhist[K_];
    __shared__ float chist[K_];

    const int wg  = blockIdx.x;
    const int tid = threadIdx.x;
    const int n0  = wg * PIX;          // PIX divides HW -> stripe stays in one image
    const int b   = n0 / HW;
    const int p0  = n0 % HW;

    for (int i = tid; i < PIX * 16; i += 128)
        ssqp[i] = 0.0f;
    if (tid < K_) { nhist[tid] = 0.0f; chist[tid] = 0.0f; }
    __syncthreads();

    const int wave = tid >> 5;
    const int lane = tid & 31;
    const int half = lane >> 4;        // 0: K=0,1 lanes ; 1: K=2,3 lanes
    const int l15  = lane & 15;        // A: channel-in-group ; B/C: class column

    const float* fbase =
        feats + ((size_t)(b * C_ + wave * 64 + l15) * HW + (size_t)p0 + (size_t)half * 2);
    const long long* ybase = ydown + ((size_t)b * HW + p0);
    const size_t cgstride = (size_t)16 * HW;

    v8f acc0[4], acc1[4];              // [channel group] x {classes 0-15, 16-18}
#pragma unroll
    for (int i = 0; i < 4; ++i) { acc0[i] = (v8f)(0.0f); acc1[i] = (v8f)(0.0f); }

    // Pipeline prologue: step 0 operands.
    v2f A[4];
#pragma unroll
    for (int i = 0; i < 4; ++i)
        A[i] = *reinterpret_cast<const v2f*>(fbase + i * cgstride);
    int la0 = (int)ybase[half * 2];
    int la1 = (int)ybase[half * 2 + 1];

    for (int g = 0; g < PIX / 32; ++g) {
        // Prefetch PFDIST groups (512B per channel plane) ahead; speculative
        // prefetch past the buffer end is silently dropped.
#pragma unroll
        for (int i = 0; i < 4; ++i)
            __builtin_prefetch(fbase + i * cgstride + g * 32 + 32 * PFDIST, 0, 1);

#pragma unroll
        for (int ss = 0; ss < 8; ++ss) {
            const int lp  = g * 32 + ss * 4;
            const int nlp = (lp + 4 < PIX) ? (lp + 4) : 0;   // last step: benign reload

            // Issue next step's loads before consuming this step's operands.
            v2f An[4];
#pragma unroll
            for (int i = 0; i < 4; ++i)
                An[i] = *reinterpret_cast<const v2f*>(fbase + i * cgstride + nlp);
            const int ln0 = (int)ybase[nlp + half * 2];
            const int ln1 = (int)ybase[nlp + half * 2 + 1];

            v2f B0, B1;
            B0.x = (la0 == l15)      ? 1.0f : 0.0f;
            B0.y = (la1 == l15)      ? 1.0f : 0.0f;
            B1.x = (la0 == l15 + 16) ? 1.0f : 0.0f;
            B1.y = (la1 == l15 + 16) ? 1.0f : 0.0f;

            float s0 = 0.0f, s1 = 0.0f;
#pragma unroll
            for (int i = 0; i < 4; ++i) {
                acc0[i] = __builtin_amdgcn_wmma_f32_16x16x4_f32(
                    false, A[i], false, B0, (short)0, acc0[i], false, false);
                acc1[i] = __builtin_amdgcn_wmma_f32_16x16x4_f32(
                    false, A[i], false, B1, (short)0, acc1[i], false, false);
                s0 = fmaf(A[i].x, A[i].x, s0);
                s1 = fmaf(A[i].y, A[i].y, s1);
            }
            // Conflict-free per-lane deposit (4 waves accumulate per address);
            // no return value -> never waited on inside the loop.
            atomicAdd(&ssqp[(lp + half * 2)     * 16 + l15], s0);
            atomicAdd(&ssqp[(lp + half * 2 + 1) * 16 + l15], s1);

#pragma unroll
            for (int i = 0; i < 4; ++i) A[i] = An[i];
            la0 = ln0;
            la1 = ln1;
        }
    }

    // Flush class-sum tiles: value(vgpr r, lane) = sums[M = r + 8*half][N = l15].
    float* wsums = ws_sums + (size_t)wg * KC;
#pragma unroll
    for (int i = 0; i < 4; ++i) {
        const int chb = wave * 64 + i * 16 + half * 8;
#pragma unroll
        for (int r = 0; r < 8; ++r)
            wsums[l15 * C_ + chb + r] = acc0[i][r];
        if (l15 < 3) {                 // classes 16..18 live in tile 1 cols 0..2
#pragma unroll
            for (int r = 0; r < 8; ++r)
                wsums[(16 + l15) * C_ + chb + r] = acc1[i][r];
        }
    }

    // Phase 2: fold ssq columns, per-pixel norm, label histogram.
    __syncthreads();
    for (int i = tid; i < PIX; i += 128) {
        float t = 0.0f;
#pragma unroll
        for (int j = 0; j < 16; ++j)
            t += ssqp[i * 16 + j];
        const float nv = sqrtf(t);
        const int   l  = (int)ybase[i];
        atomicAdd(&nhist[l], nv);
        atomicAdd(&chist[l], 1.0f);
    }
    __syncthreads();
    if (tid < K_) {
        ws_norm[(size_t)wg * K_ + tid] = nhist[tid];
        ws_cnt [(size_t)wg * K_ + tid] = chist[tid];
    }
}

// -----------------------------------------------------------------------------
// Kernel 2 (stage A): fold 1024 WG partials -> 16 chunk partials, coalesced.
//   grid = (NCHUNK, NJT) x 128 threads; lane-consecutive j => 512B rows.
// -----------------------------------------------------------------------------
__global__ __launch_bounds__(128) void seg_reduceA_kernel(
    const float* __restrict__ ws_sums,
    float* __restrict__ ws2)           // [NCHUNK][KC]
{
    const int chunk = blockIdx.x;
    const int j     = blockIdx.y * JT + threadIdx.x;
    const float* p  = ws_sums + (size_t)chunk * RCHUNK * KC + j;
    float s = 0.0f;
#pragma unroll 8
    for (int w = 0; w < RCHUNK; ++w)
        s += p[(size_t)w * KC];
    ws2[(size_t)chunk * KC + j] = s;
}

// -----------------------------------------------------------------------------
// Kernel 3 (stage B): finalize means [19][256] + norm_means [19].
// -----------------------------------------------------------------------------
__global__ __launch_bounds__(256) void seg_final_kernel(
    const float* __restrict__ ws2,
    const float* __restrict__ ws_norm,
    const float* __restrict__ ws_cnt,
    float* __restrict__ out)
{
    __shared__ float red[256];
    __shared__ float cnt_sh;
    __shared__ float nrm_sh;

    const int k = blockIdx.x;
    const int c = threadIdx.x;

    float s = 0.0f;
#pragma unroll
    for (int ch = 0; ch < NCHUNK; ++ch)
        s += ws2[(size_t)ch * KC + k * C_ + c];

    float cp = 0.0f, np = 0.0f;
    for (int wg = c; wg < NWG; wg += 256) {
        cp += ws_cnt [(size_t)wg * K_ + k];
        np += ws_norm[(size_t)wg * K_ + k];
    }

    red[c] = cp; __syncthreads();
    for (int off = 128; off > 0; off >>= 1) {
        if (c < off) red[c] += red[c + off];
        __syncthreads();
    }
    if (c == 0) cnt_sh = red[0];
    __syncthreads();

    red[c] = np; __syncthreads();
    for (int off = 128; off > 0; off >>= 1) {
        if (c < off) red[c] += red[c + off];
        __syncthreads();
    }
    if (c == 0) nrm_sh = red[0];
    __syncthreads();

    const float cnt   = cnt_sh;
    const float denom = fmaxf(cnt, 1.0f);
    out[k * C_ + c] = (cnt > 0.0f) ? (s / denom) : 0.0f;          // means [K,C]
    if (c == 0)
        out[KC + k] = (cnt > 0.0f) ? (nrm_sh / denom) : 0.0f;     // norm_means [K]
}

extern "C" void kernel_launch(void* const* d_in, const int* in_sizes, int n_in,
                              void* d_out, int out_size, void* d_ws, size_t ws_size,
                              hipStream_t stream) {
    const float*     feats = (const float*)d_in[0];
    const long long* y     = (const long long*)d_in[1];

    float* ws      = (float*)d_ws;
    float* ws_sums = ws;                                   // NWG*KC floats (~19.9 MB)
    float* ws_norm = ws_sums + (size_t)NWG * KC;           // NWG*K_ floats
    float* ws_cnt  = ws_norm + (size_t)NWG * K_;           // NWG*K_ floats
    float* ws2     = ws_cnt  + (size_t)NWG * K_;           // NCHUNK*KC floats (~311 KB)
    float* out     = (float*)d_out;                        // 19*256 + 19 floats

    hipLaunchKernelGGL(seg_partial_kernel, dim3(NWG), dim3(128), 0, stream,
                       feats, y, ws_sums, ws_norm, ws_cnt);
    hipLaunchKernelGGL(seg_reduceA_kernel, dim3(NCHUNK, NJT), dim3(128), 0, stream,
                       ws_sums, ws2);
    hipLaunchKernelGGL(seg_final_kernel, dim3(K_), dim3(256), 0, stream,
                       ws2, ws_norm, ws_cnt, out);
}